// GraphConvLayer_74741020885174
// MI455X (gfx1250) — compile-verified
//
#include <hip/hip_runtime.h>

typedef __attribute__((ext_vector_type(2))) float v2f;
typedef __attribute__((ext_vector_type(8))) float v8f;

// D = A(16x4 f32) * B(4x16 f32) + C(16x16 f32), wave32 matrix op
#define WMMA_F32(a, b, c) \
  __builtin_amdgcn_wmma_f32_16x16x4_f32(false, (a), false, (b), (short)0, (c), false, false)

__global__ __launch_bounds__(256) void zero_ws_kernel(float* ws, int n) {
  int i = blockIdx.x * blockDim.x + threadIdx.x;
  if (i < n) ws[i] = 0.0f;
}

// ---------------------------------------------------------------------------
// Edge phase: msgs = relu([edge_feat | node_feats[nbr]] @ W1p + b1p) @ W2p
// One wave32 processes a 16-edge tile with f32 WMMA (K chunks of 4).
// Loads are clamped (branch-free); scatter targets are distributed by lane
// shuffle (no redundant global index loads, no per-row waits).
// ---------------------------------------------------------------------------
__global__ __launch_bounds__(256) void edge_msg_kernel(
    const float* __restrict__ node_feats,
    const int*   __restrict__ edges,       // [2, E]: row0 = neighbour, row1 = target
    const float* __restrict__ edge_feats,  // [E, 10]
    const float* __restrict__ W1p,         // [20, 10]
    const float* __restrict__ b1p,         // [10]
    const float* __restrict__ W2p,         // [10, 10]
    float* __restrict__ seg_sum,           // [N, 10]
    float* __restrict__ cnt,               // [N]
    int E)
{
  __shared__ float tile[8 * 256];          // 16x16 f32 transpose tile per wave
  const int lane = threadIdx.x & 31;
  const int wave = threadIdx.x >> 5;
  const int row  = lane & 15;              // M for A, N for B/C/D
  const int hi   = lane >> 4;              // lane-half selects K pair / row half
  float* myTile = &tile[wave * 256];

  // --- Preload B matrices in WMMA B-layout: lane holds col n=row, k = v + 2*hi
  v2f B1[5];
  #pragma unroll
  for (int c = 0; c < 5; ++c) {
    int k0 = c * 4 + 2 * hi;
    float x = (row < 10 && k0     < 20) ? W1p[k0 * 10 + row]       : 0.0f;
    float y = (row < 10 && k0 + 1 < 20) ? W1p[(k0 + 1) * 10 + row] : 0.0f;
    B1[c] = (v2f){x, y};
  }
  v2f B2[3];
  #pragma unroll
  for (int c = 0; c < 3; ++c) {
    int k0 = c * 4 + 2 * hi;
    float x = (row < 10 && k0     < 10) ? W2p[k0 * 10 + row]       : 0.0f;
    float y = (row < 10 && k0 + 1 < 10) ? W2p[(k0 + 1) * 10 + row] : 0.0f;
    B2[c] = (v2f){x, y};
  }
  const float bias1 = (row < 10) ? b1p[row] : 0.0f;

  const int numTiles = (E + 15) >> 4;
  const int wavesPerBlock = blockDim.x >> 5;
  const int waveId   = blockIdx.x * wavesPerBlock + wave;
  const int numWaves = gridDim.x * wavesPerBlock;

  for (int t = waveId; t < numTiles; t += numWaves) {
    // prefetch next tile's streaming edge features
    int tn = t + numWaves;
    if (tn < numTiles)
      __builtin_prefetch(&edge_feats[(size_t)tn * 160 + lane * 4], 0, 1);

    const int e    = t * 16 + row;
    const int eC   = (e < E) ? e : (E - 1);        // clamp: branch-free loads
    const int nb   = edges[eC];                    // neighbour index (L2 gather)
    const int tgtL = edges[E + eC];                // this row's target node
    const float* pE = edge_feats + (size_t)eC * 10;
    const float* pN = node_feats + (size_t)nb * 10;

    // --- Layer 1: acc = A(16x20) @ W1p, 5 WMMA K-chunks of 4.
    // Each (chunk, lane-half) fragment k = {4c+2hi, 4c+2hi+1} is contiguous,
    // 8B-aligned, and entirely on one side of the concat -> single b64 load.
    v8f acc = {0.f, 0.f, 0.f, 0.f, 0.f, 0.f, 0.f, 0.f};
    #pragma unroll
    for (int c = 0; c < 5; ++c) {
      int k0 = c * 4 + 2 * hi;
      const float* p = (k0 < 10) ? (pE + k0) : (pN + (k0 - 10));
      v2f a = *(const v2f*)p;
      acc = WMMA_F32(a, B1[c], acc);
    }

    // --- bias + relu, transpose via wave-private LDS (C layout -> A layout)
    #pragma unroll
    for (int g = 0; g < 8; ++g) {
      float v = acc[g] + bias1;
      v = v > 0.0f ? v : 0.0f;             // cols >=10 are exact zeros (padded W)
      myTile[(g + 8 * hi) * 16 + row] = v;
    }

    // --- Layer 2: msgs = relu1(16x10, padded to 12) @ W2p, 3 WMMA chunks
    v8f acc2 = {0.f, 0.f, 0.f, 0.f, 0.f, 0.f, 0.f, 0.f};
    #pragma unroll
    for (int c = 0; c < 3; ++c) {
      int k0 = c * 4 + 2 * hi;
      v2f a2 = (v2f){myTile[row * 16 + k0], myTile[row * 16 + k0 + 1]};
      acc2 = WMMA_F32(a2, B2[c], acc2);
    }

    // --- Distribute the 8 target node ids this lane needs (rows m = g + 8*hi)
    // via lane permute, with all lanes active (sources are lanes 0..15).
    int tgtm[8];
    #pragma unroll
    for (int g = 0; g < 8; ++g)
      tgtm[g] = __shfl(tgtL, g + 8 * hi, 32);

    // one count atomic per edge, done by the hi==0 lane of each row
    if (hi == 0 && e < E) atomicAdd(&cnt[tgtL], 1.0f);

    // --- Scatter messages: lane holds (m = g + 8*hi, n = row)
    if (row < 10) {
      if (t * 16 + 16 <= E) {              // fast path: full tile, no compares
        #pragma unroll
        for (int g = 0; g < 8; ++g)
          atomicAdd(&seg_sum[(size_t)tgtm[g] * 10 + row], acc2[g]);
      } else {                             // tail tile
        #pragma unroll
        for (int g = 0; g < 8; ++g)
          if (t * 16 + g + 8 * hi < E)
            atomicAdd(&seg_sum[(size_t)tgtm[g] * 10 + row], acc2[g]);
      }
    }
  }
}

// ---------------------------------------------------------------------------
// Node phase: out = relu([node | seg_sum/cnt] @ W1u + b1u) @ W2u
// Layer 1 has 20 output cols -> two WMMA column tiles (0..15 and 16..19 pad).
// ---------------------------------------------------------------------------
__global__ __launch_bounds__(256) void node_update_kernel(
    const float* __restrict__ node_feats,  // [N, 10]
    const float* __restrict__ seg_sum,     // [N, 10]
    const float* __restrict__ cnt,         // [N]
    const float* __restrict__ W1u,         // [20, 20]
    const float* __restrict__ b1u,         // [20]
    const float* __restrict__ W2u,         // [20, 10]
    float* __restrict__ out,               // [N, 10]
    int N)
{
  __shared__ float tile[8 * 512];          // 16x32 f32 transpose tile per wave
  const int lane = threadIdx.x & 31;
  const int wave = threadIdx.x >> 5;
  const int row  = lane & 15;
  const int hi   = lane >> 4;
  float* myTile = &tile[wave * 512];

  v2f B1a[5], B1b[5], B2[5];
  #pragma unroll
  for (int c = 0; c < 5; ++c) {
    int k0 = c * 4 + 2 * hi;
    int k1 = k0 + 1;
    B1a[c] = (v2f){(k0 < 20) ? W1u[k0 * 20 + row] : 0.0f,
                   (k1 < 20) ? W1u[k1 * 20 + row] : 0.0f};
    int n2 = 16 + row;
    B1b[c] = (v2f){(k0 < 20 && n2 < 20) ? W1u[k0 * 20 + n2] : 0.0f,
                   (k1 < 20 && n2 < 20) ? W1u[k1 * 20 + n2] : 0.0f};
    B2[c]  = (v2f){(row < 10 && k0 < 20) ? W2u[k0 * 10 + row] : 0.0f,
                   (row < 10 && k1 < 20) ? W2u[k1 * 10 + row] : 0.0f};
  }
  const float biasA = b1u[row];                               // row < 16 < 20
  const float biasB = (16 + row < 20) ? b1u[16 + row] : 0.0f;

  const int numTiles = (N + 15) >> 4;
  const int wavesPerBlock = blockDim.x >> 5;
  const int waveId   = blockIdx.x * wavesPerBlock + wave;
  const int numWaves = gridDim.x * wavesPerBlock;

  for (int t = waveId; t < numTiles; t += numWaves) {
    const int node = t * 16 + row;
    const int nS   = (node < N) ? node : (N - 1);   // clamp: branch-free loads
    const float c0  = cnt[nS];
    const float inv = (c0 > 0.0f) ? (1.0f / c0) : 0.0f;      // segment mean
    const float* pF = node_feats + (size_t)nS * 10;
    const float* pS = seg_sum    + (size_t)nS * 10;

    v8f accA = {0.f, 0.f, 0.f, 0.f, 0.f, 0.f, 0.f, 0.f};
    v8f accB = {0.f, 0.f, 0.f, 0.f, 0.f, 0.f, 0.f, 0.f};
    #pragma unroll
    for (int c = 0; c < 5; ++c) {
      int k0 = c * 4 + 2 * hi;
      const float* p = (k0 < 10) ? (pF + k0) : (pS + (k0 - 10));
      const float  s = (k0 < 10) ? 1.0f : inv;      // mean scale, branch-free
      v2f a = *(const v2f*)p;
      a.x *= s; a.y *= s;
      accA = WMMA_F32(a, B1a[c], accA);
      accB = WMMA_F32(a, B1b[c], accB);
    }

    // bias + relu, store 16x20 (in 16x32 region) to LDS for transpose
    #pragma unroll
    for (int g = 0; g < 8; ++g) {
      int m = g + 8 * hi;
      float va = accA[g] + biasA; va = va > 0.0f ? va : 0.0f;
      float vb = accB[g] + biasB; vb = vb > 0.0f ? vb : 0.0f; // cols >=20 -> 0
      myTile[m * 32 + row]      = va;
      myTile[m * 32 + 16 + row] = vb;
    }

    // Layer 2: (16x20) @ W2u, 5 WMMA K-chunks
    v8f acc2 = {0.f, 0.f, 0.f, 0.f, 0.f, 0.f, 0.f, 0.f};
    #pragma unroll
    for (int c = 0; c < 5; ++c) {
      int k0 = c * 4 + 2 * hi;
      v2f a2 = (v2f){myTile[row * 32 + k0], myTile[row * 32 + k0 + 1]};
      acc2 = WMMA_F32(a2, B2[c], acc2);
    }

    if (row < 10) {
      const int base = t * 16;
      if (base + 16 <= N) {                // fast path: full tile
        #pragma unroll
        for (int g = 0; g < 8; ++g)
          out[(size_t)(base + g + 8 * hi) * 10 + row] = acc2[g];
      } else {                             // tail tile
        #pragma unroll
        for (int g = 0; g < 8; ++g) {
          int m = g + 8 * hi;
          if (base + m < N) out[(size_t)(base + m) * 10 + row] = acc2[g];
        }
      }
    }
  }
}

extern "C" void kernel_launch(void* const* d_in, const int* in_sizes, int n_in,
                              void* d_out, int out_size, void* d_ws, size_t ws_size,
                              hipStream_t stream) {
  const float* node_feats = (const float*)d_in[0];
  const int*   edges      = (const int*)  d_in[1];
  const float* edge_feats = (const float*)d_in[2];
  const float* W1p        = (const float*)d_in[3];
  const float* b1p        = (const float*)d_in[4];
  const float* W2p        = (const float*)d_in[5];
  const float* W1u        = (const float*)d_in[6];
  const float* b1u        = (const float*)d_in[7];
  const float* W2u        = (const float*)d_in[8];
  float* out = (float*)d_out;

  const int N = in_sizes[0] / 10;
  const int E = in_sizes[1] / 2;

  float* seg_sum = (float*)d_ws;               // N*10 floats
  float* cntbuf  = seg_sum + (size_t)N * 10;   // N floats

  // zero the scatter workspace every call (deterministic under graph replay)
  const int zn = N * 11;
  zero_ws_kernel<<<(zn + 255) / 256, 256, 0, stream>>>(seg_sum, zn);

  // edge phase: grid-stride over 16-edge tiles, 8 waves per block
  edge_msg_kernel<<<4096, 256, 0, stream>>>(node_feats, edges, edge_feats,
                                            W1p, b1p, W2p, seg_sum, cntbuf, E);

  // node phase: one wave per 16-node tile
  const int nodeTiles  = (N + 15) >> 4;
  const int nodeBlocks = (nodeTiles + 7) / 8;
  node_update_kernel<<<nodeBlocks, 256, 0, stream>>>(node_feats, seg_sum, cntbuf,
                                                     W1u, b1u, W2u, out, N);
}